// LinearAttentionBlock_64158221467761
// MI455X (gfx1250) — compile-verified
//
#include <hip/hip_runtime.h>
#include <hip/hip_bf16.h>
#include <math.h>

typedef __attribute__((ext_vector_type(2))) float v2f;
typedef __attribute__((ext_vector_type(4))) float v4f;
typedef __attribute__((ext_vector_type(8))) float v8f;

#define B_  8
#define T_  512
#define F_  512
#define D_  128
#define BT_ (B_*T_)
#define NC_ (T_/16)

// D = A(16x4, f32) * B(4x16, f32) + C(16x16, f32)   -> v_wmma_f32_16x16x4_f32
__device__ __forceinline__ v8f wmma_f32_k4(v2f a, v2f b, v8f c) {
  return __builtin_amdgcn_wmma_f32_16x16x4_f32(false, a, false, b, (short)0, c, false, false);
}

// CDNA5 async global->LDS copy (ASYNCcnt-tracked, no VGPR round trip).
// LDS byte address = low 32 bits of the generic shared-memory pointer.
__device__ __forceinline__ void async_b128(void* lds, const float* gaddr) {
  uint32_t lo = (uint32_t)(uintptr_t)lds;
  asm volatile("global_load_async_to_lds_b128 %0, %1, off"
               :: "v"(lo), "v"(gaddr) : "memory");
}
__device__ __forceinline__ void wait_async0() {
  asm volatile("s_wait_asynccnt 0" ::: "memory");
}

// ---------------- LayerNorm over F, one row per block ----------------
__global__ void ln_kernel(const float* __restrict__ x, const float* __restrict__ g,
                          const float* __restrict__ bb, float* __restrict__ xn) {
  __shared__ float red[256];
  const int row = blockIdx.x, tid = threadIdx.x;
  const float* xr = x + (size_t)row * F_;
  float x0 = xr[tid], x1 = xr[tid + 256];
  red[tid] = x0 + x1;
  __syncthreads();
  for (int s = 128; s > 0; s >>= 1) { if (tid < s) red[tid] += red[tid + s]; __syncthreads(); }
  float mu = red[0] * (1.0f / F_);
  __syncthreads();
  float d0 = x0 - mu, d1 = x1 - mu;
  red[tid] = d0 * d0 + d1 * d1;
  __syncthreads();
  for (int s = 128; s > 0; s >>= 1) { if (tid < s) red[tid] += red[tid + s]; __syncthreads(); }
  float rstd = rsqrtf(red[0] * (1.0f / F_) + 1e-5f);
  float* xo = xn + (size_t)row * F_;
  xo[tid]       = d0 * rstd * g[tid]       + bb[tid];
  xo[tid + 256] = d1 * rstd * g[tid + 256] + bb[tid + 256];
}

// -------- generic 16x16-tile WMMA GEMM: out = act(A[M,K]@W[K,128] + bias) + resid --------
// act: 0 = none, 1 = elu, 2 = relu.  One wave (32 lanes) per 16x16 tile.
__global__ void gemm16(const float* __restrict__ A, int K,
                       const float* __restrict__ W,
                       const float* __restrict__ bias,
                       const float* __restrict__ resid,
                       float* __restrict__ out, int act) {
  const int m0 = blockIdx.x * 16, n0 = blockIdx.y * 16;
  const int lane = threadIdx.x, hi = lane >> 4, lr = lane & 15;
  const float* Arow = A + (size_t)(m0 + lr) * K;
  v8f acc = {};
  for (int k = 0; k < K; k += 4) {
    const int ka = k + 2 * hi;
    v2f a; a.x = Arow[ka];                     a.y = Arow[ka + 1];
    v2f b; b.x = W[(size_t)ka * D_ + n0 + lr]; b.y = W[(size_t)(ka + 1) * D_ + n0 + lr];
    acc = wmma_f32_k4(a, b, acc);
  }
  const float bv = bias ? bias[n0 + lr] : 0.f;
  for (int r = 0; r < 8; ++r) {
    const int row = m0 + r + 8 * hi, col = n0 + lr;
    float v = acc[r] + bv;
    if (act == 1)      v = v > 0.f ? v : (expf(v) - 1.f);
    else if (act == 2) v = fmaxf(v, 0.f);
    if (resid) v += resid[(size_t)row * D_ + col];
    out[(size_t)row * D_ + col] = v;
  }
}

// -------- chunked linear-attention scan. grid = (B, D/16); block = 256 (8 waves) --------
__global__ void scan_kernel(const float* __restrict__ Kb, const float* __restrict__ Qb,
                            const float* __restrict__ Vb, const float* __restrict__ S0,
                            const float* __restrict__ Z0,
                            float* __restrict__ Pout,   // pre-MLP numerator/den  [BT,D]
                            float* __restrict__ Sout,   // S output [B,T,D,D]
                            float* __restrict__ Zout) { // Z output [B,T,D]
  __shared__ float Ssl[D_][16];              // state slice S[:, 16j:16j+16]
  __shared__ float Qc[2][16][D_], Kc[2][16][D_], Vc[2][16][D_];   // double-buffered chunks
  __shared__ float Asm[16][16];              // QK^T (unmasked)
  __shared__ float part[8][16][16];          // per-wave k-partials of numerator tile
  __shared__ float den[16];
  __shared__ float Z0s[D_];
  const int b = blockIdx.x, j = blockIdx.y;
  const int tid = threadIdx.x, wave = tid >> 5, lane = tid & 31;
  const int hi = lane >> 4, lr = lane & 15;

  for (int e = tid; e < D_ * 16; e += 256) {           // S slice init from S0[b]
    const int di = e >> 4, jj = e & 15;
    Ssl[di][jj] = S0[((size_t)b * D_ + di) * D_ + 16 * j + jj];
  }
  if (tid < D_) Z0s[tid] = Z0[(size_t)b * D_ + tid];

  // prologue: async-stage chunk 0 into buffer 0
  {
    const size_t base0 = (size_t)b * T_ * D_;
    for (int e = tid; e < 512; e += 256) {
      const int t = e >> 5, dq = (e & 31) * 4;
      async_b128(&Qc[0][t][dq], &Qb[base0 + (size_t)t * D_ + dq]);
      async_b128(&Kc[0][t][dq], &Kb[base0 + (size_t)t * D_ + dq]);
      async_b128(&Vc[0][t][dq], &Vb[base0 + (size_t)t * D_ + dq]);
    }
  }

  for (int c = 0; c < NC_; ++c) {
    const int cur = c & 1, nxt = cur ^ 1;
    const int t0 = c * 16;
    wait_async0();                                     // chunk c landed in LDS (this wave)
    __syncthreads();                                   // ... visible to all; chunk c-1 retired
    if (tid < 16) den[tid] = 0.f;
    if (c + 1 < NC_) {                                 // overlap next chunk's staging w/ compute
      const size_t basen = ((size_t)b * T_ + t0 + 16) * D_;
      for (int e = tid; e < 512; e += 256) {
        const int t = e >> 5, dq = (e & 31) * 4;
        async_b128(&Qc[nxt][t][dq], &Qb[basen + (size_t)t * D_ + dq]);
        async_b128(&Kc[nxt][t][dq], &Kb[basen + (size_t)t * D_ + dq]);
        async_b128(&Vc[nxt][t][dq], &Vb[basen + (size_t)t * D_ + dq]);
      }
    }
    __syncthreads();                                   // den reset visible

    {                                                  // den[t] = Q[t].(K[t]+Z0) ; emit Z slice
      const int t = tid >> 4, g = tid & 15;
      float p = 0.f;
      for (int d = g * 8; d < g * 8 + 8; ++d) p += Qc[cur][t][d] * (Kc[cur][t][d] + Z0s[d]);
      atomicAdd(&den[t], p);
      const int zc = 16 * j + g;
      __builtin_nontemporal_store(Kc[cur][t][zc] + Z0s[zc],
                                  &Zout[((size_t)b * T_ + t0 + t) * D_ + zc]);
    }
    if (wave == 0) {                                   // A = Q K^T  (16x16, K=128)
      v8f acc = {};
      for (int k = 0; k < D_; k += 4) {
        const int ka = k + 2 * hi;
        v2f a;  a.x = Qc[cur][lr][ka];  a.y = Qc[cur][lr][ka + 1];
        v2f bb; bb.x = Kc[cur][lr][ka]; bb.y = Kc[cur][lr][ka + 1];
        acc = wmma_f32_k4(a, bb, acc);
      }
      for (int r = 0; r < 8; ++r) Asm[r + 8 * hi][lr] = acc[r];
    }
    {                                                  // numerator tile: Q@S_prev split over waves
      v8f acc = {};
      const int kb = wave * 16;
      for (int k = kb; k < kb + 16; k += 4) {
        const int ka = k + 2 * hi;
        v2f a;  a.x = Qc[cur][lr][ka]; a.y = Qc[cur][lr][ka + 1];
        v2f bb; bb.x = Ssl[ka][lr];    bb.y = Ssl[ka + 1][lr];
        acc = wmma_f32_k4(a, bb, acc);
      }
      if (wave == 0) {                                 // + tril(A) @ V slice (causal, incl diag)
        for (int k = 0; k < 16; k += 4) {
          const int ka = k + 2 * hi;
          v2f a;
          a.x = (ka     <= lr) ? Asm[lr][ka]     : 0.f;
          a.y = (ka + 1 <= lr) ? Asm[lr][ka + 1] : 0.f;
          v2f bb; bb.x = Vc[cur][ka][16 * j + lr]; bb.y = Vc[cur][ka + 1][16 * j + lr];
          acc = wmma_f32_k4(a, bb, acc);
        }
      }
      for (int r = 0; r < 8; ++r) part[wave][r + 8 * hi][lr] = acc[r];
    }
    __syncthreads();                                   // partials/den/A done; S_sl reads done

    {                                                  // reduce partials, divide, emit pre-MLP
      const int t = tid >> 4, s = tid & 15;
      float sum = 0.f;
      for (int w = 0; w < 8; ++w) sum += part[w][t][s];
      const float d = den[t];
      const float sd = (d > 0.f) ? 1.f : ((d < 0.f) ? -1.f : 0.f);
      const float dc = sd * fmaxf(fabsf(d), 1e-8f);
      Pout[((size_t)b * T_ + t0 + t) * D_ + 16 * j + s] = sum / dc;
    }
    {                                                  // running prefix of outer(K,V): NT-stream S_t
      for (int f = tid; f < 512; f += 256) {           // 512 float4s cover the 128x16 slice
        const int di = f >> 2, jj = (f & 3) * 4;
        v4f s4 = *(v4f*)&Ssl[di][jj];
        for (int t = 0; t < 16; ++t) {
          const float kv = Kc[cur][t][di];
          s4.x += kv * Vc[cur][t][16 * j + jj + 0];
          s4.y += kv * Vc[cur][t][16 * j + jj + 1];
          s4.z += kv * Vc[cur][t][16 * j + jj + 2];
          s4.w += kv * Vc[cur][t][16 * j + jj + 3];
          __builtin_nontemporal_store(
              s4, (v4f*)&Sout[(((size_t)b * T_ + t0 + t) * D_ + di) * D_ + 16 * j + jj]);
        }
        *(v4f*)&Ssl[di][jj] = s4;                      // carry state into next chunk
      }
    }
  }
}

extern "C" void kernel_launch(void* const* d_in, const int* in_sizes, int n_in,
                              void* d_out, int out_size, void* d_ws, size_t ws_size,
                              hipStream_t stream) {
  const float* x   = (const float*)d_in[0];
  const float* Wk  = (const float*)d_in[1];
  const float* Wq  = (const float*)d_in[2];
  const float* Wv  = (const float*)d_in[3];
  const float* lng = (const float*)d_in[4];
  const float* lnb = (const float*)d_in[5];
  const float* W1  = (const float*)d_in[6];
  const float* b1  = (const float*)d_in[7];
  const float* W2  = (const float*)d_in[8];
  const float* b2  = (const float*)d_in[9];
  const float* Wsw = (const float*)d_in[10];
  const float* bs  = (const float*)d_in[11];
  const float* S0  = (const float*)d_in[12];
  const float* Z0  = (const float*)d_in[13];

  float* out  = (float*)d_out;                       // [BT,D]
  float* Sout = out + (size_t)BT_ * D_;              // [B,T,D,D]
  float* Zout = Sout + (size_t)BT_ * D_ * D_;        // [BT,D]

  float* ws = (float*)d_ws;
  float* xn = ws;                                    // BT*F
  float* Kb = xn + (size_t)BT_ * F_;                 // each BT*D below
  float* Qb = Kb + (size_t)BT_ * D_;
  float* Vb = Qb + (size_t)BT_ * D_;
  float* Rb = Vb + (size_t)BT_ * D_;
  float* Pb = Rb + (size_t)BT_ * D_;
  float* Hb = Pb + (size_t)BT_ * D_;

  ln_kernel<<<BT_, 256, 0, stream>>>(x, lng, lnb, xn);

  dim3 gg(BT_ / 16, D_ / 16);
  gemm16<<<gg, 32, 0, stream>>>(xn, F_, Wk,  nullptr, nullptr, Kb, 1);   // K = elu(xn@Wk)
  gemm16<<<gg, 32, 0, stream>>>(xn, F_, Wq,  nullptr, nullptr, Qb, 1);   // Q = elu(xn@Wq)
  gemm16<<<gg, 32, 0, stream>>>(xn, F_, Wv,  nullptr, nullptr, Vb, 0);   // V = xn@Wv
  gemm16<<<gg, 32, 0, stream>>>(xn, F_, Wsw, bs,      nullptr, Rb, 0);   // R = xn@Ws + bs

  scan_kernel<<<dim3(B_, D_ / 16), 256, 0, stream>>>(Kb, Qb, Vb, S0, Z0, Pb, Sout, Zout);

  gemm16<<<gg, 32, 0, stream>>>(Pb, D_, W1, b1, nullptr, Hb, 2);         // H = relu(P@W1+b1)
  gemm16<<<gg, 32, 0, stream>>>(Hb, D_, W2, b2, Rb,      out, 2);        // out = relu(H@W2+b2)+R
}